// MS_MLP_v1_74792560493051
// MI455X (gfx1250) — compile-verified
//
#include <hip/hip_runtime.h>
#include <math.h>

// MS-MLP block for MI455X (gfx1250), fp32 end-to-end using V_WMMA_F32_16X16X4_F32,
// with TDM (TENSOR_LOAD_TO_LDS) staging of shared A/weight panels into LDS.
// B=8, C=64, H=W=256, P=4.

#define BBATCH 8
#define CCH    64
#define HDIM   256
#define WDIM   256
#define HWSZ   65536L

typedef __attribute__((ext_vector_type(2))) float v2f;
typedef __attribute__((ext_vector_type(8))) float v8f;
typedef __attribute__((ext_vector_type(4))) unsigned int v4u;
typedef __attribute__((ext_vector_type(8))) int v8i;
typedef __attribute__((ext_vector_type(4))) int v4i;

#if __has_builtin(__builtin_amdgcn_tensor_load_to_lds)
#define HAVE_TDM 1
#else
#define HAVE_TDM 0
#endif

__device__ __forceinline__ v8f wmma_f32x4(v2f a, v2f b, v8f c) {
  // D(16x16,f32) = A(16x4,f32) * B(4x16,f32) + C
  return __builtin_amdgcn_wmma_f32_16x16x4_f32(
      /*neg_a=*/false, a, /*neg_b=*/false, b,
      /*c_mod=*/(short)0, c, /*reuse_a=*/false, /*reuse_b=*/false);
}

__device__ __forceinline__ float gelu_exact(float x) {
  return 0.5f * x * (1.0f + erff(x * 0.70710678118654752f));
}

#if HAVE_TDM
// 1-D linear TDM copy: n fp32 elements (n <= 16384) from global to LDS.
// D# per CDNA5 ISA 8.3/8.4: group0 = {count=1, lds_addr, global_addr[56:0], type=2},
// group1 = {data_size=4B, tensor_dim0=n, tensor_dim1=1, tile_dim0=n, tile_dim1=1,
//           tensor_dim0_stride=n}; groups 2/3 zero (<=2D tensor).
// This toolchain exposes the 6-arg builtin: (v4u, v8i, v4i, v4i, v8i, cpol).
__device__ __forceinline__ void tdm_load_lin(void* lds_dst, const void* gsrc,
                                             unsigned n) {
  const unsigned long long ga = (unsigned long long)gsrc;
  const unsigned lds_off = (unsigned)(unsigned long long)lds_dst;  // addr[31:0] = LDS offset
  v4u g0;
  g0.x = 1u;                                        // count=1, user descriptor
  g0.y = lds_off;                                   // lds_addr (bytes)
  g0.z = (unsigned)(ga & 0xffffffffu);              // global_addr[31:0] -> bits 95:64
  g0.w = (unsigned)((ga >> 32) & 0x01ffffffu) | 0x80000000u;  // addr[56:32] | type=2
  v8i g1;
  g1[0] = 0x20000;                                  // data_size=2 (4 bytes)
  g1[1] = (int)((n & 0xffffu) << 16);               // tensor_dim0[15:0] @ bit48
  g1[2] = (int)(((n >> 16) & 0xffffu) | (1u << 16));// tensor_dim0[31:16] | tensor_dim1=1
  g1[3] = (int)((n & 0xffffu) << 16);               // tile_dim0 = n (16-bit)
  g1[4] = 1;                                        // tile_dim1 = 1 (tile_dim2 = 0)
  g1[5] = (int)n;                                   // tensor_dim0_stride[31:0]
  g1[6] = 0;
  g1[7] = 0;
  v4i z4 = {0, 0, 0, 0};
  v8i z8 = {0, 0, 0, 0, 0, 0, 0, 0};
  __builtin_amdgcn_tensor_load_to_lds(g0, g1, z4, z4, z8, 0);
}
#endif

// Block-cooperative stage of n fp32 (n <= 16384) into LDS, safe for reuse of the
// LDS buffer across loop iterations (barrier before overwrite + after fill).
__device__ __forceinline__ void stage_lds(float* dst, const float* src, int n) {
  __syncthreads();                         // all waves done reading previous panel
#if HAVE_TDM
  if ((threadIdx.x >> 5) == 0) {           // one wave issues the DMA (EXEC ignored)
    tdm_load_lin(dst, src, (unsigned)n);
    __builtin_amdgcn_s_wait_tensorcnt(0);  // DMA landed in LDS
  }
#else
  for (int i = threadIdx.x; i < n; i += 256) dst[i] = src[i];
#endif
  __syncthreads();
}

// ---------------------------------------------------------------------------
// Fused channel-LayerNorm (over C=64) + 1x1 conv (OC = 16*nOT), optional GELU.
// One wave = 16 consecutive pixels of one batch image. LDS stages 64x16 tile.
// ---------------------------------------------------------------------------
__global__ __launch_bounds__(256) void k_ln_mm(
    const float* __restrict__ in, const float* __restrict__ lnw,
    const float* __restrict__ lnb, const float* __restrict__ w,
    int nOT, int act, float* __restrict__ out, long outBStr) {
  __shared__ float lds[8 * CCH * 16];
  const int wid = threadIdx.x >> 5;
  const int lane = threadIdx.x & 31;
  const int l16 = lane & 15;
  const int hi = lane >> 4;
  const long task = (long)blockIdx.x * 8 + wid;   // NPIX/16 tasks
  const long b = task >> 12;                      // 4096 groups per image
  const long sp0 = (task & 4095) * 16;
  const float* xb = in + b * (CCH * HWSZ) + sp0;
  float* lw = &lds[wid * CCH * 16];

  for (int i = lane; i < CCH * 16; i += 32)
    lw[i] = xb[(long)(i >> 4) * HWSZ + (i & 15)];
  __syncthreads();

  if (lane < 16) {  // per-pixel stats over 64 channels (biased var)
    float mu = 0.f, s2 = 0.f;
    for (int c = 0; c < CCH; ++c) {
      float v = lw[c * 16 + lane];
      mu += v; s2 += v * v;
    }
    mu *= (1.0f / CCH);
    s2 = s2 * (1.0f / CCH) - mu * mu;
    float inv = rsqrtf(s2 + 1e-6f);
    for (int c = 0; c < CCH; ++c)
      lw[c * 16 + lane] = (lw[c * 16 + lane] - mu) * inv * lnw[c] + lnb[c];
  }
  __syncthreads();

  for (int ot = 0; ot < nOT; ++ot) {
    v8f acc = {};
#pragma unroll
    for (int k = 0; k < 16; ++k) {        // K = 64
      const int k0 = k * 4 + 2 * hi;
      v2f a, bb;
      a.x = w[(ot * 16 + l16) * CCH + k0];
      a.y = w[(ot * 16 + l16) * CCH + k0 + 1];
      bb.x = lw[k0 * 16 + l16];
      bb.y = lw[(k0 + 1) * 16 + l16];
      acc = wmma_f32x4(a, bb, acc);
    }
#pragma unroll
    for (int v = 0; v < 8; ++v) {
      const int o = ot * 16 + v + 8 * hi;
      float r = acc[v];
      if (act) r = gelu_exact(r);
      out[b * outBStr + (long)o * HWSZ + sp0 + l16] = r;
    }
  }
}

// ---------------------------------------------------------------------------
// Patch-linear: 4x4 patches, per (patch, 16-channel tile): D(c,s) = In(c,t) W(s,t)^T + bias.
// Exactly one 16x16 WMMA tile (K=16 -> 4 steps). Optional fused gate multiply.
// ---------------------------------------------------------------------------
__global__ __launch_bounds__(256) void k_patch_linear(
    const float* __restrict__ in, long inB,
    const float* __restrict__ w, const float* __restrict__ bias,
    const float* __restrict__ gate, long gB,
    float* __restrict__ out, long outB) {
  const int wid = threadIdx.x >> 5;
  const int lane = threadIdx.x & 31;
  const int l16 = lane & 15;
  const int hi = lane >> 4;
  const long task = (long)blockIdx.x * 8 + wid;   // patches(32768) * ctiles(4)
  const int ct = (int)(task & 3);
  const long n = task >> 2;
  const long b = n >> 12;
  const int hp = (int)((n >> 6) & 63);
  const int wp = (int)(n & 63);
  const long cbase = (long)(ct * 16) * HWSZ + (long)hp * 4 * WDIM + wp * 4;
  const float* ip = in + b * inB + cbase;

  v8f acc = {};
#pragma unroll
  for (int k = 0; k < 4; ++k) {           // K = 16 spatial taps
    const int t0 = k * 4 + 2 * hi;
    const int t1 = t0 + 1;
    v2f a, bb;
    a.x = ip[(long)l16 * HWSZ + (t0 >> 2) * WDIM + (t0 & 3)];
    a.y = ip[(long)l16 * HWSZ + (t1 >> 2) * WDIM + (t1 & 3)];
    bb.x = w[l16 * 16 + t0];              // B[t][s] = lin_w[s][t]
    bb.y = w[l16 * 16 + t1];
    acc = wmma_f32x4(a, bb, acc);
  }
#pragma unroll
  for (int v = 0; v < 8; ++v) {
    const int c = v + 8 * hi;
    const int s = l16;
    const long off = (long)c * HWSZ + (s >> 2) * WDIM + (s & 3);
    float r = acc[v] + bias[s];
    if (gate) r *= gate[b * gB + cbase + off];
    out[b * outB + cbase + off] = r;
  }
}

// ---------------------------------------------------------------------------
// Spatial GEMM: per (b,c) plane, D(256x256) = A(256x256) * B(256x256).
// One wave = one 16x16 tile, K=256 -> 64 WMMA steps. All 8 waves in a block
// share the same A row-panel (16x256 fp32, contiguous 16KB) -> TDM-stage it
// into LDS once per block; B streams from global (the HBM-bound operand).
// ---------------------------------------------------------------------------
__global__ __launch_bounds__(256) void k_sgemm(
    const float* __restrict__ A, long aB, long aC,
    const float* __restrict__ Bm, long bB, long bC,
    const float* __restrict__ gate, long gB,
    float* __restrict__ out) {
  __shared__ float As[16 * 256];                  // 16KB A panel
  const int wid = threadIdx.x >> 5;
  const int lane = threadIdx.x & 31;
  const int l16 = lane & 15;
  const int hi = lane >> 4;
  const long task = (long)blockIdx.x * 8 + wid;   // 512 planes * 256 tiles
  const int st = (int)(task & 15);
  const int pt = (int)((task >> 4) & 15);         // block-uniform
  const long bc = task >> 8;                      // block-uniform
  const long bi = bc >> 6;
  const long ci = bc & 63;
  const float* Ap = A + bi * aB + ci * aC + (long)pt * 16 * 256;  // panel start
  const float* Bb = Bm + bi * bB + ci * bC;

  stage_lds(As, Ap, 16 * 256);

  v8f acc = {};
#pragma unroll 8
  for (int k = 0; k < 64; ++k) {          // K = 256
    const int k0 = k * 4 + 2 * hi;
    v2f a, bb;
    a.x = As[l16 * 256 + k0];
    a.y = As[l16 * 256 + k0 + 1];
    bb.x = Bb[(long)k0 * 256 + st * 16 + l16];
    bb.y = Bb[(long)(k0 + 1) * 256 + st * 16 + l16];
    acc = wmma_f32x4(a, bb, acc);
  }
  const long obase = bi * (CCH * HWSZ) + ci * HWSZ;
#pragma unroll
  for (int v = 0; v < 8; ++v) {
    const int p = pt * 16 + v + 8 * hi;
    const long off = (long)p * 256 + st * 16 + l16;
    float r = acc[v];
    if (gate) r *= gate[bi * gB + ci * HWSZ + off];
    out[obase + off] = r;
  }
}

// ---------------------------------------------------------------------------
// fc2 over concat(g1,g2) (K=128) + residual x0. Weight strips (16x128) are
// block-uniform -> TDM-stage per o-tile.
// ---------------------------------------------------------------------------
__global__ __launch_bounds__(256) void k_fc2_res(
    const float* __restrict__ g1, const float* __restrict__ g2,
    const float* __restrict__ w, const float* __restrict__ res,
    float* __restrict__ out) {
  __shared__ float Ws[16 * 128];                  // 8KB weight strip
  const int wid = threadIdx.x >> 5;
  const int lane = threadIdx.x & 31;
  const int l16 = lane & 15;
  const int hi = lane >> 4;
  const long task = (long)blockIdx.x * 8 + wid;
  const long b = task >> 12;
  const long sp0 = (task & 4095) * 16;
  const float* p1 = g1 + b * (CCH * HWSZ) + sp0;
  const float* p2 = g2 + b * (CCH * HWSZ) + sp0;

  for (int ot = 0; ot < 4; ++ot) {
    stage_lds(Ws, w + (long)ot * 16 * 128, 16 * 128);
    v8f acc = {};
#pragma unroll
    for (int k = 0; k < 32; ++k) {        // K = 128 (k0 even -> no half-crossing)
      const int k0 = k * 4 + 2 * hi;
      v2f a, bb;
      a.x = Ws[l16 * 128 + k0];
      a.y = Ws[l16 * 128 + k0 + 1];
      const float* src = (k0 < 64) ? p1 + (long)k0 * HWSZ
                                   : p2 + (long)(k0 - 64) * HWSZ;
      bb.x = src[l16];
      bb.y = src[HWSZ + l16];
      acc = wmma_f32x4(a, bb, acc);
    }
#pragma unroll
    for (int v = 0; v < 8; ++v) {
      const int o = ot * 16 + v + 8 * hi;
      const long addr = b * (CCH * HWSZ) + (long)o * HWSZ + sp0 + l16;
      out[addr] = acc[v] + res[addr];
    }
  }
}

// ---------------------------------------------------------------------------
// Depthwise 3x3 (SAME, zero pad) + exact GELU. Bandwidth-bound -> plain VALU.
// ---------------------------------------------------------------------------
__global__ __launch_bounds__(256) void k_dwconv_gelu(
    const float* __restrict__ in, const float* __restrict__ kw,
    float* __restrict__ out) {
  const long idx = (long)blockIdx.x * 256 + threadIdx.x;  // B*256*HW
  const int wx = (int)(idx & 255);
  const int hx = (int)((idx >> 8) & 255);
  const long cb = idx >> 16;
  const int ch = (int)(cb & 255);
  const float* kp = kw + ch * 9;
  const float* ip = in + (cb << 16);
  float s = 0.f;
#pragma unroll
  for (int kh = 0; kh < 3; ++kh) {
#pragma unroll
    for (int kx = 0; kx < 3; ++kx) {
      const int h2 = hx + kh - 1;
      const int w2 = wx + kx - 1;
      if (h2 >= 0 && h2 < 256 && w2 >= 0 && w2 < 256)
        s = fmaf(kp[kh * 3 + kx], ip[h2 * 256 + w2], s);
    }
  }
  out[idx] = gelu_exact(s);
}

// ---------------------------------------------------------------------------
// FFN second 1x1 (K=256) + residual -> final output. TDM-staged weight strips.
// ---------------------------------------------------------------------------
__global__ __launch_bounds__(256) void k_ffn2_res(
    const float* __restrict__ y, const float* __restrict__ w,
    const float* __restrict__ res, float* __restrict__ out) {
  __shared__ float Ws[16 * 256];                  // 16KB weight strip
  const int wid = threadIdx.x >> 5;
  const int lane = threadIdx.x & 31;
  const int l16 = lane & 15;
  const int hi = lane >> 4;
  const long task = (long)blockIdx.x * 8 + wid;
  const long b = task >> 12;
  const long sp0 = (task & 4095) * 16;
  const float* yb = y + b * (256 * HWSZ) + sp0;

  for (int ot = 0; ot < 4; ++ot) {
    stage_lds(Ws, w + (long)ot * 16 * 256, 16 * 256);
    v8f acc = {};
#pragma unroll 8
    for (int k = 0; k < 64; ++k) {        // K = 256
      const int k0 = k * 4 + 2 * hi;
      v2f a, bb;
      a.x = Ws[l16 * 256 + k0];
      a.y = Ws[l16 * 256 + k0 + 1];
      const float* src = yb + (long)k0 * HWSZ;
      bb.x = src[l16];
      bb.y = src[HWSZ + l16];
      acc = wmma_f32x4(a, bb, acc);
    }
#pragma unroll
    for (int v = 0; v < 8; ++v) {
      const int o = ot * 16 + v + 8 * hi;
      const long addr = b * (CCH * HWSZ) + (long)o * HWSZ + sp0 + l16;
      out[addr] = acc[v] + res[addr];
    }
  }
}

// ---------------------------------------------------------------------------
extern "C" void kernel_launch(void* const* d_in, const int* in_sizes, int n_in,
                              void* d_out, int out_size, void* d_ws, size_t ws_size,
                              hipStream_t stream) {
  (void)in_sizes; (void)n_in; (void)out_size; (void)ws_size;
  const float* x0       = (const float*)d_in[0];
  const float* m1       = (const float*)d_in[1];
  const float* m2       = (const float*)d_in[2];
  const float* norm_w   = (const float*)d_in[3];
  const float* norm_b   = (const float*)d_in[4];
  const float* fc1_w    = (const float*)d_in[5];
  const float* lin1_w   = (const float*)d_in[6];
  const float* lin1_b   = (const float*)d_in[7];
  const float* lin2_w   = (const float*)d_in[8];
  const float* lin2_b   = (const float*)d_in[9];
  const float* fc2_w    = (const float*)d_in[10];
  const float* ffn_ln_w = (const float*)d_in[11];
  const float* ffn_ln_b = (const float*)d_in[12];
  const float* ffn_w1   = (const float*)d_in[13];
  const float* ffn_dw   = (const float*)d_in[14];
  const float* ffn_w2   = (const float*)d_in[15];
  float* out = (float*)d_out;
  float* ws  = (float*)d_ws;

  // workspace layout (floats)
  float* X2 = ws;                          // (B,128,HW) fc1 output
  float* T1 = X2 + 8L * 128 * HWSZ;        // (B,64,HW)  x_g1
  float* T2 = T1 + 8L * 64 * HWSZ;         // (B,64,HW)  rotating temp
  float* T3 = T2 + 8L * 64 * HWSZ;         // (B,64,HW)  x4 then xout1
  float* Y1 = T3 + 8L * 64 * HWSZ;         // (B,256,HW) FFN hidden (+sgemm scratch)
  float* Y2 = Y1 + 8L * 256 * HWSZ;        // (B,256,HW) after dw-conv

  const long s64  = 64L * HWSZ;
  const long s128 = 128L * HWSZ;
  const long s256 = 256L * HWSZ;
  const float* X2h = X2 + 64L * HWSZ;      // x2_2 base (channels 64..127)

  dim3 blk(256);
  // 1) LN2d + fc1 (64 -> 128)
  hipLaunchKernelGGL(k_ln_mm, dim3(4096), blk, 0, stream,
                     x0, norm_w, norm_b, fc1_w, 8, 0, X2, s128);
  // 2) branch 1: patch-linear lin1 with fused gate (x_g1 = x2_1 * x3) -> T1
  hipLaunchKernelGGL(k_patch_linear, dim3(16384), blk, 0, stream,
                     X2, s128, lin1_w, lin1_b, X2, s128, T1, s64);
  // 3) branch 2: U = m1 @ x2_2 -> T2
  hipLaunchKernelGGL(k_sgemm, dim3(16384), blk, 0, stream,
                     m1, 0L, 0L, X2h, s128, HWSZ, (const float*)nullptr, 0L, T2);
  // 4) x4 = U @ m2 -> T3
  hipLaunchKernelGGL(k_sgemm, dim3(16384), blk, 0, stream,
                     T2, s64, HWSZ, m2, 0L, 0L, (const float*)nullptr, 0L, T3);
  // 5) x5 = patch-linear lin2 on x4 -> T2
  hipLaunchKernelGGL(k_patch_linear, dim3(16384), blk, 0, stream,
                     T3, s64, lin2_w, lin2_b, (const float*)nullptr, 0L, T2, s64);
  // 6) U2 = m1 @ x5 -> Y1 (scratch region, overwritten later by FFN)
  hipLaunchKernelGGL(k_sgemm, dim3(16384), blk, 0, stream,
                     m1, 0L, 0L, T2, s64, HWSZ, (const float*)nullptr, 0L, Y1);
  // 7) x_g2 = x2_2 * (U2 @ m2) -> T2 (fused gate)
  hipLaunchKernelGGL(k_sgemm, dim3(16384), blk, 0, stream,
                     Y1, s64, HWSZ, m2, 0L, 0L, X2h, s128, T2);
  // 8) xout1 = fc2(concat(x_g1, x_g2)) + x0 -> T3
  hipLaunchKernelGGL(k_fc2_res, dim3(4096), blk, 0, stream,
                     T1, T2, fc2_w, x0, T3);
  // 9) FFN: LN + w1 (64 -> 256) + GELU -> Y1
  hipLaunchKernelGGL(k_ln_mm, dim3(4096), blk, 0, stream,
                     T3, ffn_ln_w, ffn_ln_b, ffn_w1, 16, 1, Y1, s256);
  // 10) depthwise 3x3 + GELU -> Y2
  hipLaunchKernelGGL(k_dwconv_gelu, dim3(524288), blk, 0, stream,
                     Y1, ffn_dw, Y2);
  // 11) w2 (256 -> 64) + residual xout1 -> out
  hipLaunchKernelGGL(k_ffn2_res, dim3(4096), blk, 0, stream,
                     Y2, ffn_w2, T3, out);
  // 12) tuple tail: m1, m2 pass-through
  (void)hipMemcpyAsync(out + 8L * 64 * HWSZ, m1, 65536 * sizeof(float),
                       hipMemcpyDeviceToDevice, stream);
  (void)hipMemcpyAsync(out + 8L * 64 * HWSZ + 65536, m2, 65536 * sizeof(float),
                       hipMemcpyDeviceToDevice, stream);
}